// GraphConv_88364657147964
// MI455X (gfx1250) — compile-verified
//
#include <hip/hip_runtime.h>
#include <hip/hip_bf16.h>

// ---------------------------------------------------------------------------
// GraphConv for MI455X (gfx1250, wave32):
//   h   = x @ W.T + b              (f32 WMMA 16x16x4)
//   out = relu(rownorm(A+I) @ h)   (L2-resident gather + f32 atomics scatter)
// ---------------------------------------------------------------------------

#define D_FEAT 128

typedef float v2f __attribute__((ext_vector_type(2)));
typedef float v8f __attribute__((ext_vector_type(8)));

// ---- degree kernels --------------------------------------------------------

__global__ void gc_init_counts(unsigned* __restrict__ cnt, int n) {
    int i = blockIdx.x * blockDim.x + threadIdx.x;
    if (i < n) cnt[i] = 1u;               // self-loop
}

__global__ void gc_count_edges(const int* __restrict__ erow,
                               unsigned* __restrict__ cnt, int ne) {
    int e = blockIdx.x * blockDim.x + threadIdx.x;
    if (e < ne) atomicAdd(&cnt[erow[e]], 1u);
}

__global__ void gc_inv_deg(const unsigned* __restrict__ cnt,
                           float* __restrict__ invd, int n) {
    int i = blockIdx.x * blockDim.x + threadIdx.x;
    if (i < n) invd[i] = 1.0f / (float)cnt[i];
}

// ---- GEMM: h = x @ W.T + b  via V_WMMA_F32_16X16X4_F32 ---------------------
// grid.x = M/16 tiles (M % 16 == 0), blockDim = 256 (8 waves),
// wave w computes N-tile w (N = 128 = 8 * 16). K = 128 = 32 * 4 steps.
//
// ISA operand layouts (CDNA5 ISA, 32-bit 16x4 A):
//   lane L: Mrow = L%16, Kpair = 2*(L/16); VGPR0/1 = A[Mrow][Kpair+0/1]
// B (4x16 = W.T tile) mirrors: lane L holds {B[Kpair][N=L%16], B[Kpair+1][N]}
//   = {W[n0+L%16][k+Kpair], W[n0+L%16][k+Kpair+1]}  -> contiguous float2 of W.
// C/D: VGPR v, lane L -> M = v + 8*(L/16), N = L%16.

__global__ void gc_gemm_wmma(const float* __restrict__ x,
                             const float* __restrict__ W,
                             const float* __restrict__ bias,
                             float* __restrict__ h) {
    const int lane  = threadIdx.x & 31;
    const int wave  = threadIdx.x >> 5;          // n-tile 0..7
    const int m0    = blockIdx.x << 4;
    const int n0    = wave << 4;
    const int r16   = lane & 15;
    const int khalf = (lane >> 4) << 1;          // 0 or 2

    const float* aptr = x + (m0 + r16) * D_FEAT + khalf;
    const float* bptr = W + (n0 + r16) * D_FEAT + khalf;

    v8f acc = {};
#pragma unroll
    for (int k = 0; k < D_FEAT; k += 4) {
        v2f a = *(const v2f*)(aptr + k);
        v2f b = *(const v2f*)(bptr + k);
        acc = __builtin_amdgcn_wmma_f32_16x16x4_f32(
            /*neg_a=*/false, a, /*neg_b=*/false, b,
            /*c_mod=*/(short)0, acc, /*reuse_a=*/false, /*reuse_b=*/false);
    }

    const float bv    = bias[n0 + r16];
    const int   mbase = m0 + ((lane >> 4) << 3); // +8 for upper half lanes
    const int   ncol  = n0 + r16;
#pragma unroll
    for (int v = 0; v < 8; ++v)
        h[(mbase + v) * D_FEAT + ncol] = acc[v] + bv;
}

// ---- self-loop init: out[i,:] = h[i,:] * inv_deg[i] ------------------------
// one lane per 4 dims: n * 32 threads total

__global__ void gc_self_loop(const float* __restrict__ h,
                             const float* __restrict__ invd,
                             float* __restrict__ out, int n) {
    int gid = blockIdx.x * blockDim.x + threadIdx.x;
    if (gid >= n * 32) return;
    int i = gid >> 5;
    int q = (gid & 31) << 2;
    float c = invd[i];
    float4 hv = *(const float4*)(h + i * D_FEAT + q);
    float4 o  = make_float4(c * hv.x, c * hv.y, c * hv.z, c * hv.w);
    *(float4*)(out + i * D_FEAT + q) = o;
}

// ---- edge scatter: out[row,:] += inv_deg[row] * h[col,:] -------------------
// one wave (32 lanes x float4) per edge; 4 f32 atomics per lane.
// h and out are both L2-resident (51 MB each vs 192 MB L2).

__global__ void gc_scatter(const float* __restrict__ h,
                           const int* __restrict__ erow,
                           const int* __restrict__ ecol,
                           const float* __restrict__ invd,
                           float* __restrict__ out, int ne) {
    long long gid = (long long)blockIdx.x * blockDim.x + threadIdx.x;
    if (gid >= (long long)ne * 32) return;
    int e = (int)(gid >> 5);
    int q = ((int)gid & 31) << 2;
    int r = erow[e];
    int c = ecol[e];
    float coef = invd[r];
    float4 hv  = *(const float4*)(h + c * D_FEAT + q);
    float* o   = out + r * D_FEAT + q;
    atomicAdd(o + 0, coef * hv.x);
    atomicAdd(o + 1, coef * hv.y);
    atomicAdd(o + 2, coef * hv.z);
    atomicAdd(o + 3, coef * hv.w);
}

// ---- ReLU ------------------------------------------------------------------

__global__ void gc_relu(float* __restrict__ out, int n4) {
    int i = blockIdx.x * blockDim.x + threadIdx.x;
    if (i >= n4) return;
    float4 v = *(const float4*)(out + (i << 2));
    v.x = fmaxf(v.x, 0.0f);
    v.y = fmaxf(v.y, 0.0f);
    v.z = fmaxf(v.z, 0.0f);
    v.w = fmaxf(v.w, 0.0f);
    *(float4*)(out + (i << 2)) = v;
}

// ---------------------------------------------------------------------------

extern "C" void kernel_launch(void* const* d_in, const int* in_sizes, int n_in,
                              void* d_out, int out_size, void* d_ws, size_t ws_size,
                              hipStream_t stream) {
    const float* x    = (const float*)d_in[0];   // [N, 128]
    const float* W    = (const float*)d_in[1];   // [128, 128]
    const float* bias = (const float*)d_in[2];   // [128]
    const int*   er   = (const int*)d_in[3];     // [NE]
    const int*   ec   = (const int*)d_in[4];     // [NE]
    float*       out  = (float*)d_out;           // [N, 128]

    const int N  = in_sizes[0] / D_FEAT;         // 100000
    const int NE = in_sizes[3];                  // 1600000

    // workspace: h [N*128 f32] | counts [N u32] | inv_deg [N f32]
    float*    h    = (float*)d_ws;
    unsigned* cnt  = (unsigned*)(h + (size_t)N * D_FEAT);
    float*    invd = (float*)(cnt + N);

    const int B = 256;

    // 1) degree (with self-loop) and its reciprocal
    gc_init_counts<<<(N + B - 1) / B, B, 0, stream>>>(cnt, N);
    gc_count_edges<<<(NE + B - 1) / B, B, 0, stream>>>(er, cnt, NE);
    gc_inv_deg<<<(N + B - 1) / B, B, 0, stream>>>(cnt, invd, N);

    // 2) h = x @ W.T + b   (N % 16 == 0 -> exact tiling, 8 waves/block)
    gc_gemm_wmma<<<N / 16, B, 0, stream>>>(x, W, bias, h);

    // 3) out = h * inv_deg (self-loop contribution initializes the accumulator)
    gc_self_loop<<<(N * 32 + B - 1) / B, B, 0, stream>>>(h, invd, out, N);

    // 4) out[row] += inv_deg[row] * h[col] over all edges
    long long sthreads = (long long)NE * 32;
    gc_scatter<<<(unsigned)((sthreads + B - 1) / B), B, 0, stream>>>(h, er, ec, invd, out, NE);

    // 5) out = relu(out)
    int n4 = N * D_FEAT / 4;
    gc_relu<<<(n4 + B - 1) / B, B, 0, stream>>>(out, n4);
}